// Attention_26895085207604
// MI455X (gfx1250) — compile-verified
//
#include <hip/hip_runtime.h>
#include <hip/hip_bf16.h>

#define B_ 16
#define T_ 4096
#define E_ 300

typedef __attribute__((ext_vector_type(2))) float v2f;
typedef __attribute__((ext_vector_type(8))) float v8f;

__device__ __forceinline__ float fast_sigmoid(float x) {
    // 1 / (1 + 2^(-x*log2(e)))
    return __builtin_amdgcn_rcpf(1.0f + __builtin_amdgcn_exp2f(-1.4426950408889634f * x));
}
__device__ __forceinline__ float fast_tanh(float x) {
    // 2*sigmoid(2x) - 1
    return 2.0f * __builtin_amdgcn_rcpf(1.0f + __builtin_amdgcn_exp2f(-2.8853900817779268f * x)) - 1.0f;
}

// ---------------------------------------------------------------------------
// K1: pre-gate GEMM via V_WMMA_F32_16X16X4_F32.
// Tile: M = 16 consecutive timesteps of one batch, N = 16 cols
//       (n<4: fwd gates i,f,g,o ; 4<=n<8: bwd gates ; n>=8: zero pad),
// K loop over E=300 in steps of 4.
// Output layout: gates[t][dir*16 + b][gate]  (float, so 128 floats per t)
// ---------------------------------------------------------------------------
__global__ __launch_bounds__(256) void gate_gemm(
    const float* __restrict__ x,       // [B,T,E]
    const float* __restrict__ Wih_f,   // [4,E]
    const float* __restrict__ Wih_b,   // [4,E]
    const float* __restrict__ b_f,     // [4]
    const float* __restrict__ b_b,     // [4]
    float* __restrict__ gates)         // [T,32,4]
{
    const int wave = threadIdx.x >> 5;           // 8 waves per block
    const int lane = threadIdx.x & 31;
    const int tile = blockIdx.x * 8 + wave;      // 4096 tiles total
    const int b    = tile >> 8;                  // T/16 = 256 tiles per batch
    const int t0   = (tile & 255) << 4;

    const int m     = lane & 15;                 // A-matrix row within tile
    const int khalf = (lane >> 4) << 1;          // 0 for lanes 0-15, 2 for 16-31
    const int n     = lane & 15;                 // B/D column
    const int g     = n & 3;

    // Branchless weight-row selection: lanes with n>=8 read a valid row but
    // multiply by 0, keeping EXEC all-ones around the WMMA.
    const float* __restrict__ Wp = (n < 4) ? (Wih_f + g * E_) : (Wih_b + g * E_);
    const float wmask = (n < 8) ? 1.0f : 0.0f;

    const float* __restrict__ xrow = x + ((size_t)b * T_ + (t0 + m)) * E_;

    v8f acc = {0.f, 0.f, 0.f, 0.f, 0.f, 0.f, 0.f, 0.f};

    for (int k0 = 0; k0 < E_; k0 += 4) {
        // A: 16x4 fp32 (lanes 0-15: K0 in v0, K1 in v1; lanes 16-31: K2, K3)
        v2f a = *(const v2f*)(xrow + k0 + khalf);
        // B: 4x16 fp32, mirrored layout (lane-half selects K pair)
        v2f bmat;
        bmat.x = wmask * Wp[k0 + khalf];
        bmat.y = wmask * Wp[k0 + khalf + 1];
        acc = __builtin_amdgcn_wmma_f32_16x16x4_f32(
            /*neg_a=*/false, a, /*neg_b=*/false, bmat,
            /*c_mod=*/(short)0, acc, /*reuse_a=*/false, /*reuse_b=*/false);
    }

    // D layout: VGPR r -> M = r + 8*(lane>>4), N = lane&15.
    if (n < 8) {
        const int dir   = n >> 2;
        const float bias = (dir == 0) ? b_f[g] : b_b[g];
        const int mbase = (lane >> 4) << 3;
        const int slot  = dir * 16 + b;
#pragma unroll
        for (int r = 0; r < 8; ++r) {
            const int t = t0 + mbase + r;
            gates[(size_t)t * 128 + slot * 4 + g] = acc[r] + bias;
        }
    }
}

// ---------------------------------------------------------------------------
// K2: the sequential recurrence. 32 chains == one wave32.
// lane = dir*16 + b. Forward lanes walk t ascending, backward descending.
// ---------------------------------------------------------------------------
__global__ __launch_bounds__(32) void lstm_scan(
    const float* __restrict__ gates,   // [T,32,4]
    const float* __restrict__ Whh_f,   // [4,1]
    const float* __restrict__ Whh_b,   // [4,1]
    float* __restrict__ hbuf)          // [32,T]  (slot-major)
{
    const int lane = threadIdx.x;
    const int dir  = lane >> 4;
    const float* __restrict__ Whh = dir ? Whh_b : Whh_f;
    const float wi = Whh[0], wf = Whh[1], wg = Whh[2], wo = Whh[3];

    float h = 0.0f, c = 0.0f;
    for (int t = 0; t < T_; ++t) {
        const int tt = dir ? (T_ - 1 - t) : t;
        const float4 z = *(const float4*)(gates + (size_t)tt * 128 + lane * 4);
        const float zi = z.x + wi * h;
        const float zf = z.y + wf * h;
        const float zg = z.z + wg * h;
        const float zo = z.w + wo * h;
        const float ig = fast_sigmoid(zi);
        const float fg = fast_sigmoid(zf);
        const float gg = fast_tanh(zg);
        const float og = fast_sigmoid(zo);
        c = fg * c + ig * gg;
        h = og * fast_tanh(c);
        hbuf[(size_t)lane * T_ + tt] = h;
    }
}

// ---------------------------------------------------------------------------
// K3: masked softmax over time, one block per batch.
// scores[b][t] = 0.5*(h_fwd + h_bwd), -inf where t >= len.
// ---------------------------------------------------------------------------
__global__ __launch_bounds__(256) void softmax_attn(
    const float* __restrict__ hbuf,    // [32,T]
    const int* __restrict__ lens,      // [B]
    float* __restrict__ attn)          // [B,T]
{
    const int b   = blockIdx.x;
    const int tid = threadIdx.x;
    const int len = lens[b];
    const float NEG_INF = -__builtin_inff();

    __shared__ float sm[256];

    float s[16];
    float lmax = NEG_INF;
#pragma unroll
    for (int j = 0; j < 16; ++j) {
        const int t = tid + j * 256;
        const float v = 0.5f * (hbuf[(size_t)b * T_ + t] + hbuf[(size_t)(16 + b) * T_ + t]);
        s[j] = (t < len) ? v : NEG_INF;
        lmax = fmaxf(lmax, s[j]);
    }
    sm[tid] = lmax;
    __syncthreads();
    for (int off = 128; off > 0; off >>= 1) {
        if (tid < off) sm[tid] = fmaxf(sm[tid], sm[tid + off]);
        __syncthreads();
    }
    const float mx = sm[0];
    __syncthreads();

    float e[16];
    float lsum = 0.0f;
#pragma unroll
    for (int j = 0; j < 16; ++j) {
        // exp2(-inf) == 0 handles masked slots.
        e[j] = __builtin_amdgcn_exp2f((s[j] - mx) * 1.4426950408889634f);
        lsum += e[j];
    }
    sm[tid] = lsum;
    __syncthreads();
    for (int off = 128; off > 0; off >>= 1) {
        if (tid < off) sm[tid] += sm[tid + off];
        __syncthreads();
    }
    const float inv = __builtin_amdgcn_rcpf(sm[0]);

#pragma unroll
    for (int j = 0; j < 16; ++j) {
        const int t = tid + j * 256;
        attn[(size_t)b * T_ + t] = e[j] * inv;
    }
}

// ---------------------------------------------------------------------------
// K4: out[b,t,e] = attn[b,t] * x[b,t,e]   (float4 vectorized; E=300 % 4 == 0)
// ---------------------------------------------------------------------------
__global__ __launch_bounds__(256) void scale_out(
    const float* __restrict__ x,
    const float* __restrict__ attn,
    float* __restrict__ out)
{
    const size_t i = (size_t)blockIdx.x * blockDim.x + threadIdx.x; // over N/4
    const float4 xv = ((const float4*)x)[i];
    const size_t row = (i * 4) / E_;        // = b*T + t (rows are 4-aligned)
    const float a = attn[row];
    float4 ov;
    ov.x = xv.x * a; ov.y = xv.y * a; ov.z = xv.z * a; ov.w = xv.w * a;
    ((float4*)out)[i] = ov;
}

extern "C" void kernel_launch(void* const* d_in, const int* in_sizes, int n_in,
                              void* d_out, int out_size, void* d_ws, size_t ws_size,
                              hipStream_t stream) {
    (void)in_sizes; (void)n_in; (void)out_size; (void)ws_size;
    const float* x     = (const float*)d_in[0];
    const int*   lens  = (const int*)d_in[1];
    const float* Wih_f = (const float*)d_in[2];
    const float* Whh_f = (const float*)d_in[3];
    const float* b_f   = (const float*)d_in[4];
    const float* Wih_b = (const float*)d_in[5];
    const float* Whh_b = (const float*)d_in[6];
    const float* b_b   = (const float*)d_in[7];
    float* out = (float*)d_out;

    // Workspace layout (floats): gates[T*128] | hbuf[32*T] | attn[B*T]
    float* gates = (float*)d_ws;                       // 2 MB
    float* hbuf  = gates + (size_t)T_ * 128;           // 512 KB
    float* attn  = hbuf + (size_t)32 * T_;             // 256 KB

    // K1: 4096 tiles, 8 waves (tiles) per 256-thread block -> 512 blocks
    gate_gemm<<<dim3(512), dim3(256), 0, stream>>>(x, Wih_f, Wih_b, b_f, b_b, gates);
    // K2: the serial scan — exactly one wave32
    lstm_scan<<<dim3(1), dim3(32), 0, stream>>>(gates, Whh_f, Whh_b, hbuf);
    // K3: softmax per batch
    softmax_attn<<<dim3(B_), dim3(256), 0, stream>>>(hbuf, lens, attn);
    // K4: scale: (B*T*E)/4 = 4,915,200 float4 elems / 256 = 19200 blocks
    scale_out<<<dim3(19200), dim3(256), 0, stream>>>(x, attn, out);
}